// GatSymmetryReduce_1451698946384
// MI455X (gfx1250) — compile-verified
//
#include <hip/hip_runtime.h>
#include <stdint.h>

// GAT-style softmax-weighted neighbor reduce.
// N=50000 nodes, K=32 mails/node, F=128 features.
// Memory-bound (~870 MB @ 23.3 TB/s ~= 37us); strategy: one wave32 per node,
// DMA the node's 16KB ft tile into LDS with GLOBAL_LOAD_ASYNC_TO_LDS_B128
// (ASYNCcnt), overlap with softmax weight computation, then weighted-reduce
// from LDS. WMMA is inapplicable: the contraction carries the batch index on
// both operands (per-node 1x32 @ 32x128 vec-mat), so there is no shared matrix.

#define KDIM 32
#define FDIM 128
#define WPB 4           // waves per block -> 4 * 16 KB = 64 KB LDS / block
#define NEG_SLOPE 0.01f

// Issue KROW async row copies sharing ONE lds-address VGPR and ONE global
// address VGPR pair: the 24-bit instruction offset is added to BOTH the LDS
// and global addresses (CDNA5 ISA), and our per-row stride
// is 512 B on both sides. Template recursion guarantees literal immediates.
template <int KROW>
struct IssueRows {
  static __device__ __forceinline__ void run(uint32_t l, const float* g) {
    IssueRows<KROW - 1>::run(l, g);
    asm volatile("global_load_async_to_lds_b128 %0, %1, off offset:%2"
                 :
                 : "v"(l), "v"(g), "n"((KROW - 1) * FDIM * 4)
                 : "memory");
  }
};
template <>
struct IssueRows<0> {
  static __device__ __forceinline__ void run(uint32_t, const float*) {}
};

__device__ __forceinline__ void wait_async_le(int n) {
  if (n == 0)
    asm volatile("s_wait_asynccnt 0" ::: "memory");
  else if (n == 16)
    asm volatile("s_wait_asynccnt 16" ::: "memory");
}

extern "C" __global__ void __launch_bounds__(WPB * 32)
gat_softmax_reduce(const float* __restrict__ a1_node,
                   const float* __restrict__ a2_node,
                   const float* __restrict__ a1_mail,
                   const float* __restrict__ a2_mail,
                   const float* __restrict__ ft,
                   float* __restrict__ out,
                   int n_nodes) {
  __shared__ float tile[WPB][KDIM * FDIM];  // 16 KB per wave

  const int lane = threadIdx.x & 31;
  const int wave = threadIdx.x >> 5;
  const int n = blockIdx.x * WPB + wave;
  if (n >= n_nodes) return;  // uniform per wave

  // ---- kick off the async DMA of ft[n, :, :] into this wave's LDS tile ----
  // One b128/lane x 32 lanes = 512 B = exactly one row ft[n, k, :].
  const float* g = ft + (size_t)n * (KDIM * FDIM) + lane * 4;
  // generic LDS pointer: low 32 bits are the DS byte offset
  const uint32_t l = (uint32_t)(uintptr_t)&tile[wave][lane * 4];
  IssueRows<KDIM>::run(l, g);

  // ---- softmax weights (overlapped with the DMA), lane == k ----
  const float a1n = a1_node[n];
  const float a2n = a2_node[n];
  const float a1m = a1_mail[(size_t)n * KDIM + lane];
  const float a2m = a2_mail[(size_t)n * KDIM + lane];

  // a + b = a1_node + a2_mail + 2*(a2_node + a1_mail)
  float x = a1n + a2m + 2.0f * (a2n + a1m);
  x = (x > 0.0f) ? x : NEG_SLOPE * x;  // leaky_relu

  float m = x;
#pragma unroll
  for (int off = 16; off > 0; off >>= 1)
    m = fmaxf(m, __shfl_xor(m, off, 32));
  const float e = __expf(x - m);
  float s = e;
#pragma unroll
  for (int off = 16; off > 0; off >>= 1)
    s += __shfl_xor(s, off, 32);
  const float w = e / s;  // softmax weight for k == lane

  // ---- staged waits: async loads complete in order, so asynccnt<=16 means
  // rows 0..15 have landed; reduce those while rows 16..31 are in flight. ----
  float4 acc = make_float4(0.f, 0.f, 0.f, 0.f);

  wait_async_le(16);
#pragma unroll
  for (int k = 0; k < KDIM / 2; ++k) {
    // constant-lane broadcast via SALU/VALU readlane: no LDS traffic
    const float wk =
        __int_as_float(__builtin_amdgcn_readlane(__float_as_int(w), k));
    const float4 r = *(const float4*)&tile[wave][k * FDIM + lane * 4];
    acc.x = fmaf(wk, r.x, acc.x);
    acc.y = fmaf(wk, r.y, acc.y);
    acc.z = fmaf(wk, r.z, acc.z);
    acc.w = fmaf(wk, r.w, acc.w);
  }

  wait_async_le(0);
#pragma unroll
  for (int k = KDIM / 2; k < KDIM; ++k) {
    const float wk =
        __int_as_float(__builtin_amdgcn_readlane(__float_as_int(w), k));
    const float4 r = *(const float4*)&tile[wave][k * FDIM + lane * 4];
    acc.x = fmaf(wk, r.x, acc.x);
    acc.y = fmaf(wk, r.y, acc.y);
    acc.z = fmaf(wk, r.z, acc.z);
    acc.w = fmaf(wk, r.w, acc.w);
  }

  // lane owns features [lane*4, lane*4+4)
  *(float4*)&out[(size_t)n * FDIM + lane * 4] = acc;
}

extern "C" void kernel_launch(void* const* d_in, const int* in_sizes, int n_in,
                              void* d_out, int out_size, void* d_ws,
                              size_t ws_size, hipStream_t stream) {
  const float* a1_node = (const float*)d_in[0];
  const float* a2_node = (const float*)d_in[1];
  const float* a1_mail = (const float*)d_in[2];
  const float* a2_mail = (const float*)d_in[3];
  const float* ft      = (const float*)d_in[4];
  float* out = (float*)d_out;

  const int n_nodes = in_sizes[0];  // a1_node has N elements
  const int blocks = (n_nodes + WPB - 1) / WPB;

  hipLaunchKernelGGL(gat_softmax_reduce, dim3(blocks), dim3(WPB * 32), 0,
                     stream, a1_node, a2_node, a1_mail, a2_mail, ft, out,
                     n_nodes);
}